// EfnDet_50869592655378
// MI455X (gfx1250) — compile-verified
//
#include <hip/hip_runtime.h>

typedef __attribute__((ext_vector_type(2))) float v2f;
typedef __attribute__((ext_vector_type(8))) float v8f;

#define NBIN   1024
#define CAP    4096      // candidate buffer (u64 keys) for pre-NMS survivors
#define KMAX   100       // MAX_RET
#define KPAD   112       // 100 padded to 7x16 for WMMA tiles
#define CMAX   90        // NUM_CLASSES

// ---------------------------------------------------------------------------
// Kernel 1: one workgroup per (image, class).
//  Pass A: histogram of scores > 0.3 into 1024 bins -> threshold bin T so that
//          the collected set is a superset of the true top-100.
//  Pass B: collect survivors as u64 keys (score_bits<<32 | ~index), bitonic
//          sort descending in LDS, keep top-100.
//  Then:  gather anchors+deltas, regress+clip, build 112x112 suppression
//         matrix M[j][i] = (j<i) & (iou>0.45), and solve greedy NMS as the
//         fixed point  keep = valid & (M^T keep == 0)  with V_WMMA_F32_16X16X4_F32
//         on wave 0 (all 32 lanes active, as WMMA requires). LDS ordering for
//         the keep vector comes from workgroup barriers (NOT volatile, which
//         would lower to serialized flat/SYS loads).
// ---------------------------------------------------------------------------
__global__ __launch_bounds__(256)
void efndet_perclass_kernel(const float* __restrict__ bboxes,
                            const float* __restrict__ scores,
                            const float* __restrict__ anchors,
                            float* __restrict__ ws,
                            int B, int N, int C)
{
    const int bc  = blockIdx.x;
    const int b   = bc / C;
    const int c   = bc % C;
    const int tid = threadIdx.x;

    // big aliased region: phase A = hist(4KB)+cand(32KB); phase B = M (50,176B)
    __shared__ unsigned long long sbig[6272];
    unsigned int*       hist = (unsigned int*)sbig;          // [1024]
    unsigned long long* cand = sbig + 512;                   // [4096] u64
    float*              M    = (float*)sbig;                 // [112*112]

    __shared__ float4 cbox[KPAD];
    __shared__ float  cscore[KPAD];
    __shared__ float  vld[KPAD];
    __shared__ float  kcur[KPAD];
    __shared__ float  knew[KPAD];
    __shared__ unsigned long long topk[KMAX];
    __shared__ unsigned int cnt;
    __shared__ int Tbin;
    __shared__ int chg;

    for (int i = tid; i < NBIN; i += 256) hist[i] = 0u;
    if (tid == 0) cnt = 0u;
    __syncthreads();

    // ---- Pass A: histogram (strided by C; L2-resident after first touch) ----
    const float* sc = scores + (size_t)b * N * C + c;
    for (int n = tid; n < N; n += 256) {
        if (n + 2048 < N) __builtin_prefetch(sc + (size_t)(n + 2048) * C, 0, 1);
        float x = sc[(size_t)n * C];
        if (x > 0.3f) {
            int bin = (int)(x * (float)NBIN);
            bin = min(max(bin, 0), NBIN - 1);
            atomicAdd(&hist[bin], 1u);
        }
    }
    __syncthreads();

    if (tid == 0) {
        unsigned int cum = 0; int T = 0;
        for (int bin = NBIN - 1; bin >= 0; --bin) {
            cum += hist[bin];
            if (cum >= KMAX) { T = bin; break; }
        }
        Tbin = T;            // if fewer than 100 survivors total: T=0 -> take all
    }
    __syncthreads();
    const int T = Tbin;

    // ---- Pass B: collect survivors (set is deterministic; order fixed by sort) ----
    for (int n = tid; n < N; n += 256) {
        float x = sc[(size_t)n * C];
        if (x > 0.3f) {
            int bin = (int)(x * (float)NBIN);
            bin = min(max(bin, 0), NBIN - 1);
            if (bin >= T) {
                unsigned int p = atomicAdd(&cnt, 1u);
                if (p < CAP) {
                    unsigned int fb = __float_as_uint(x);   // positive floats: bit order == value order
                    cand[p] = ((unsigned long long)fb << 32) |
                              (unsigned long long)(0xFFFFFFFFu - (unsigned int)n);
                }
            }
        }
    }
    __syncthreads();
    unsigned int nc = cnt; if (nc > CAP) nc = CAP;
    for (int i = tid; i < CAP; i += 256) if (i >= (int)nc) cand[i] = 0ull;
    __syncthreads();

    // ---- bitonic sort, descending (keys unique except zero pads) ----
    for (int k = 2; k <= CAP; k <<= 1) {
        for (int j = k >> 1; j > 0; j >>= 1) {
            for (int i = tid; i < CAP; i += 256) {
                int ixj = i ^ j;
                if (ixj > i) {
                    unsigned long long a = cand[i], bb = cand[ixj];
                    bool dir = ((i & k) == 0);
                    if (dir ? (a < bb) : (a > bb)) { cand[i] = bb; cand[ixj] = a; }
                }
            }
            __syncthreads();
        }
    }
    if (tid < KMAX) topk[tid] = cand[tid];
    __syncthreads();

    // ---- gather + regress + clip top-100 boxes ----
    if (tid < KPAD) {
        float4 bx; bx.x = 0.f; bx.y = 0.f; bx.z = 0.f; bx.w = 0.f;
        float sv = -1.0f;
        if (tid < KMAX) {
            unsigned long long key = topk[tid];
            if (key != 0ull) {
                unsigned int n = 0xFFFFFFFFu - (unsigned int)(key & 0xFFFFFFFFull);
                sv = __uint_as_float((unsigned int)(key >> 32));
                const float* a4 = anchors + (size_t)n * 4;
                const float* d4 = bboxes + ((size_t)b * N + n) * 4;
                float ax1 = a4[0], ay1 = a4[1], ax2 = a4[2], ay2 = a4[3];
                float aw = ax2 - ax1, ah = ay2 - ay1;
                float acx = ax1 + 0.5f * aw, acy = ay1 + 0.5f * ah;
                float dx = d4[0] * 0.1f, dy = d4[1] * 0.1f;
                float dw = fminf(d4[2] * 0.2f, 4.135f);
                float dh = fminf(d4[3] * 0.2f, 4.135f);
                float cx = acx + dx * aw, cy = acy + dy * ah;
                float w = aw * expf(dw), h = ah * expf(dh);
                bx.x = fminf(fmaxf(cx - 0.5f * w, 0.f), 512.f);
                bx.y = fminf(fmaxf(cy - 0.5f * h, 0.f), 512.f);
                bx.z = fminf(fmaxf(cx + 0.5f * w, 0.f), 512.f);
                bx.w = fminf(fmaxf(cy + 0.5f * h, 0.f), 512.f);
            }
        }
        cbox[tid]   = bx;
        cscore[tid] = sv;
        vld[tid]    = (tid < KMAX && sv > 0.f) ? 1.f : 0.f;
        kcur[tid]   = vld[tid];
    }
    __syncthreads();

    // ---- build suppression matrix M[j][i] = (j<i) & (iou(j,i) > 0.45) ----
    for (int p = tid; p < KPAD * KPAD; p += 256) {
        int j = p / KPAD, i = p % KPAD;
        float v = 0.f;
        if (j < i && i < KMAX) {
            float4 bi = cbox[i], bj = cbox[j];
            float ai = (bi.z - bi.x) * (bi.w - bi.y);
            float aj = (bj.z - bj.x) * (bj.w - bj.y);
            float xx1 = fmaxf(bi.x, bj.x), yy1 = fmaxf(bi.y, bj.y);
            float xx2 = fminf(bi.z, bj.z), yy2 = fminf(bi.w, bj.w);
            float iw = fmaxf(xx2 - xx1, 0.f), ih = fmaxf(yy2 - yy1, 0.f);
            float inter = iw * ih;
            float iou = inter / (ai + aj - inter + 1e-8f);
            v = (iou > 0.45f) ? 1.f : 0.f;
        }
        M[p] = v;
    }
    __syncthreads();

    // ---- greedy NMS as WMMA fixed point; all waves iterate together so that
    //      plain LDS + __syncthreads provides ordering (no volatile/flat) ----
    {
        const int lane = tid & 31;
        const int half = (lane >> 4) << 1;        // 0 or 2 (A K-pair per lane half)
        const int trow = lane >> 4;               // B row within VGPR
        const int lcol = lane & 15;

        for (int it = 0; it < KMAX; ++it) {
            if (tid == 0) chg = 0;
            __syncthreads();

            if (tid < 32) {                       // wave 0: fully active for WMMA
                for (int cb = 0; cb < 7; ++cb) {
                    v8f acc = {};
                    const int col = cb * 16 + lcol;
                    for (int ch = 0; ch < 28; ++ch) {
                        const int c4 = ch * 4;
                        v2f av, bv;
                        av.x = kcur[c4 + half];
                        av.y = kcur[c4 + half + 1];
                        bv.x = M[(c4 + trow) * KPAD + col];
                        bv.y = M[(c4 + 2 + trow) * KPAD + col];
                        // D = A(16x4) * B(4x16) + C : all A-rows identical ->
                        // every D row holds r[col] = sum_j keep[j]*M[j][col]
                        acc = __builtin_amdgcn_wmma_f32_16x16x4_f32(
                            false, av, false, bv, (short)0, acc, false, false);
                    }
                    float r = acc[0];             // lanes 0-15: D[0][col]
                    if (lane < 16) {
                        int i = cb * 16 + lane;
                        knew[i] = (vld[i] > 0.5f && r < 0.5f) ? 1.f : 0.f;
                    }
                }
            }
            __syncthreads();

            if (tid < KPAD) {
                if (knew[tid] != kcur[tid]) atomicOr(&chg, 1);
                kcur[tid] = knew[tid];
            }
            __syncthreads();

            if (chg == 0) break;                  // converged (deterministic in inputs)
        }
    }

    // ---- per-class result: boxes[100*4] then scores[100] ----
    float* out = ws + (size_t)bc * 500;
    if (tid < KMAX) {
        float4 bx = cbox[tid];
        out[tid * 4 + 0] = bx.x;
        out[tid * 4 + 1] = bx.y;
        out[tid * 4 + 2] = bx.z;
        out[tid * 4 + 3] = bx.w;
        out[400 + tid]   = (kcur[tid] > 0.5f) ? cscore[tid] : -1.0f;
    }
}

// ---------------------------------------------------------------------------
// Kernel 2: per-image merge. 100 sequential argmax passes over C*100 (=9000)
// candidates held in LDS; tie-break = smaller flat index (matches lax.top_k).
// ---------------------------------------------------------------------------
__global__ __launch_bounds__(256)
void efndet_merge_kernel(const float* __restrict__ ws,
                         float* __restrict__ out_boxes,
                         float* __restrict__ out_scores,
                         float* __restrict__ out_labels,
                         int B, int C)
{
    const int b   = blockIdx.x;
    const int tid = threadIdx.x;
    const int TOT = C * KMAX;                       // 9000

    __shared__ float s[CMAX * KMAX];
    __shared__ float pv[256];
    __shared__ int   pi[256];
    __shared__ int   wsel;
    __shared__ float wscore;

    const float* base = ws + (size_t)b * C * 500;
    for (int i = tid; i < TOT; i += 256) {
        int c = i / KMAX, kk = i - c * KMAX;
        s[i] = base[(size_t)c * 500 + 400 + kk];
    }
    __syncthreads();

    for (int slot = 0; slot < KMAX; ++slot) {
        float best = -1e30f; int bi = TOT;
        for (int i = tid; i < TOT; i += 256) {
            float v = s[i];
            if (v > best) { best = v; bi = i; }
        }
        pv[tid] = best; pi[tid] = bi;
        __syncthreads();
        for (int off = 128; off > 0; off >>= 1) {
            if (tid < off) {
                float v2 = pv[tid + off]; int i2 = pi[tid + off];
                if (v2 > pv[tid] || (v2 == pv[tid] && i2 < pi[tid])) {
                    pv[tid] = v2; pi[tid] = i2;
                }
            }
            __syncthreads();
        }
        if (tid == 0) {
            wsel = pi[0]; wscore = pv[0];
            s[pi[0]] = -1e30f;                      // remove winner
        }
        __syncthreads();
        if (tid == 0) {
            int i = wsel; float sv = wscore;
            bool valid = sv > 0.f;
            int c = i / KMAX, kk = i - c * KMAX;
            const float* bp = base + (size_t)c * 500 + kk * 4;
            size_t o = ((size_t)b * KMAX + slot);
            out_boxes[o * 4 + 0] = valid ? bp[0] : 0.f;
            out_boxes[o * 4 + 1] = valid ? bp[1] : 0.f;
            out_boxes[o * 4 + 2] = valid ? bp[2] : 0.f;
            out_boxes[o * 4 + 3] = valid ? bp[3] : 0.f;
            out_scores[o] = valid ? sv : 0.f;
            out_labels[o] = valid ? (float)c : -1.0f;
        }
        __syncthreads();
    }
}

extern "C" void kernel_launch(void* const* d_in, const int* in_sizes, int n_in,
                              void* d_out, int out_size, void* d_ws, size_t ws_size,
                              hipStream_t stream)
{
    const float* bboxes  = (const float*)d_in[0];   // [B,N,4]
    const float* scores  = (const float*)d_in[1];   // [B,N,C]
    const float* anchors = (const float*)d_in[2];   // [N,4]

    const int N = in_sizes[2] / 4;                  // 49104
    const int B = in_sizes[0] / (N * 4);            // 8
    const int C = (int)((size_t)in_sizes[1] / ((size_t)B * N)); // 90

    float* ws   = (float*)d_ws;                     // B*C*500 floats = 1.44 MB
    float* outb = (float*)d_out;                    // [B,100,4]
    float* outs = outb + (size_t)B * KMAX * 4;      // [B,100]
    float* outl = outs + (size_t)B * KMAX;          // [B,100] (labels as float)

    efndet_perclass_kernel<<<B * C, 256, 0, stream>>>(bboxes, scores, anchors, ws, B, N, C);
    efndet_merge_kernel<<<B, 256, 0, stream>>>(ws, outb, outs, outl, B, C);
}